// GATEncoder_76364518523429
// MI455X (gfx1250) — compile-verified
//
#include <hip/hip_runtime.h>
#include <hip/hip_bf16.h>
#include <math.h>

typedef __attribute__((ext_vector_type(2))) float v2f;
typedef __attribute__((ext_vector_type(8))) float v8f;

#define NEG_SLOPE 0.2f

// ---------------------------------------------------------------------------
// Specialized GEMM: C[M,NN] = op(A)[M,KK] @ B[KK,NN] via V_WMMA_F32_16X16X4_F32
// Compile-time NN/KK -> immediate-offset loads, 32-bit addressing.
// One wave per 16x16 tile; K stepped by 4; relies on 192MB L2 for A/B re-reads.
// A (32-bit, 16x4): lane=16*half+m; v0=K(k0+2h), v1=K(k0+2h+1)  (float2 load).
// B (4x16): lane=16*half+n; v0 -> K=k0+2h, v1 -> K=k0+2h+1.
// C/D (16x16 f32): VGPR r, lane 16*half+n -> row (r + 8*half), col n.
// ---------------------------------------------------------------------------
template <bool RELU_A, int NN, int KK>
__global__ __launch_bounds__(256) void gemm_wmma_fix(
    const float* __restrict__ A, const float* __restrict__ B,
    float* __restrict__ C, int M) {
  const int wid = blockIdx.x * 8 + (threadIdx.x >> 5);
  constexpr int ntile = NN / 16;
  const int tm = wid / ntile;
  const int tn = wid - tm * ntile;
  if (tm * 16 >= M) return;  // uniform per-wave exit: EXEC stays all-ones

  const int lane = threadIdx.x & 31;
  const int half = lane >> 4;
  const int mr   = lane & 15;

  const float* __restrict__ arow = A + (tm * 16 + mr) * KK + 2 * half;
  const float* __restrict__ bcol = B + (2 * half) * NN + tn * 16 + mr;

  v8f acc = {};
#pragma unroll 8
  for (int k0 = 0; k0 < KK; k0 += 4) {
    v2f a = *(const v2f*)(arow + k0);  // 8B-aligned: KK,k0 multiples of 4
    if (RELU_A) { a.x = fmaxf(a.x, 0.0f); a.y = fmaxf(a.y, 0.0f); }
    v2f b;
    b.x = bcol[k0 * NN];
    b.y = bcol[k0 * NN + NN];
    // 8 args: (neg_a, A, neg_b, B, c_mod, C, reuse_a, reuse_b)
    acc = __builtin_amdgcn_wmma_f32_16x16x4_f32(
        false, a, false, b, (short)0, acc, false, false);
  }

  float* __restrict__ crow = C + (tm * 16 + 8 * half) * NN + tn * 16 + mr;
#pragma unroll
  for (int r = 0; r < 8; ++r) crow[r * NN] = acc[r];
}

// Generic fallback (runtime dims), same layout.
template <bool RELU_A>
__global__ __launch_bounds__(256) void gemm_wmma_gen(
    const float* __restrict__ A, const float* __restrict__ B,
    float* __restrict__ C, int M, int N, int K) {
  const int wid   = blockIdx.x * 8 + (threadIdx.x >> 5);
  const int ntile = N >> 4;
  const int tm = wid / ntile;
  const int tn = wid - tm * ntile;
  if (tm * 16 >= M) return;
  const int lane = threadIdx.x & 31;
  const int half = lane >> 4;
  const int mr   = lane & 15;
  const float* __restrict__ arow = A + (tm * 16 + mr) * K + 2 * half;
  const float* __restrict__ bcol = B + (2 * half) * N + tn * 16 + mr;
  v8f acc = {};
  for (int k0 = 0; k0 < K; k0 += 4) {
    v2f a = *(const v2f*)(arow + k0);
    if (RELU_A) { a.x = fmaxf(a.x, 0.0f); a.y = fmaxf(a.y, 0.0f); }
    v2f b;
    b.x = bcol[k0 * N];
    b.y = bcol[k0 * N + N];
    acc = __builtin_amdgcn_wmma_f32_16x16x4_f32(
        false, a, false, b, (short)0, acc, false, false);
  }
  float* __restrict__ crow = C + (tm * 16 + 8 * half) * N + tn * 16 + mr;
#pragma unroll
  for (int r = 0; r < 8; ++r) crow[r * N] = acc[r];
}

// ---------------------------------------------------------------------------
// Per-node attention projections: s[n]=h[n,:].a_src ; d[n]=h[n,:].a_dst
// One wave per node, lane-strided loads, wave32 shuffle reduction.
// ---------------------------------------------------------------------------
__global__ __launch_bounds__(256) void node_proj(
    const float* __restrict__ h, const float* __restrict__ asrc,
    const float* __restrict__ adst, float* __restrict__ s,
    float* __restrict__ d, int N, int F) {
  const int wid = blockIdx.x * 8 + (threadIdx.x >> 5);
  if (wid >= N) return;
  const int lane = threadIdx.x & 31;
  const float* __restrict__ hr = h + wid * F;
  float ss = 0.0f, dd = 0.0f;
  for (int f = lane; f < F; f += 32) {
    const float v = hr[f];
    ss = fmaf(v, asrc[f], ss);
    dd = fmaf(v, adst[f], dd);
  }
#pragma unroll
  for (int o = 16; o > 0; o >>= 1) {
    ss += __shfl_down(ss, o, 32);
    dd += __shfl_down(dd, o, 32);
  }
  if (lane == 0) { s[wid] = ss; d[wid] = dd; }
}

__global__ void init_node(float* __restrict__ mx, float* __restrict__ z, int N) {
  const int i = blockIdx.x * blockDim.x + threadIdx.x;
  if (i < N) { mx[i] = -INFINITY; z[i] = 0.0f; }
}

// Order-preserving float atomic max (init must be -inf).
__device__ __forceinline__ void atomicMaxFloat(float* addr, float v) {
  if (v >= 0.0f)
    atomicMax((int*)addr, __float_as_int(v));
  else
    atomicMin((unsigned int*)addr, __float_as_uint(v));
}

__global__ void edge_max_k(const int* __restrict__ ei,
                           const float* __restrict__ s,
                           const float* __restrict__ d,
                           float* __restrict__ eb, float* __restrict__ mx,
                           int E, int Etot) {
  const int e = blockIdx.x * blockDim.x + threadIdx.x;
  if (e >= Etot) return;
  int sn, dn;
  if (e < E) { sn = ei[e]; dn = ei[E + e]; } else { sn = dn = e - E; }
  float v = s[sn] + d[dn];
  v = (v >= 0.0f) ? v : NEG_SLOPE * v;  // leaky relu
  eb[e] = v;
  atomicMaxFloat(mx + dn, v);
}

__global__ void edge_exp_k(const int* __restrict__ ei,
                           const float* __restrict__ eb,
                           const float* __restrict__ mx,
                           float* __restrict__ xb, float* __restrict__ z,
                           int E, int Etot) {
  const int e = blockIdx.x * blockDim.x + threadIdx.x;
  if (e >= Etot) return;
  const int dn = (e < E) ? ei[E + e] : (e - E);
  const float ex = __expf(eb[e] - mx[dn]);
  xb[e] = ex;
  atomicAdd(z + dn, ex);
}

// out[n,f] = bias[f]  (F is a power of two)
__global__ void init_out_k(float* __restrict__ out, const float* __restrict__ b,
                           int total, int F) {
  const int i = blockIdx.x * blockDim.x + threadIdx.x;
  if (i < total) out[i] = b[i & (F - 1)];
}

// ---------------------------------------------------------------------------
// out[dst,:] += (ex[e]/z[dst]) * h[src,:] — one wave per edge, lane-strided,
// fully-unrolled immediate-offset loads + global_atomic_add_f32.
// This is the bandwidth-critical kernel (~1.3GB gathered + 218M atomics/pass).
// ---------------------------------------------------------------------------
template <int FF>
__global__ __launch_bounds__(256) void aggregate_fix(
    const int* __restrict__ ei, const float* __restrict__ xb,
    const float* __restrict__ z, const float* __restrict__ h,
    float* __restrict__ out, int E, int Etot) {
  const int wid = blockIdx.x * 8 + (threadIdx.x >> 5);
  if (wid >= Etot) return;
  const int lane = threadIdx.x & 31;
  int sn, dn;
  if (wid < E) { sn = ei[wid]; dn = ei[E + wid]; } else { sn = dn = wid - E; }
  const float coef = xb[wid] / z[dn];
  const float* __restrict__ hs = h + sn * FF + lane;
  float* __restrict__ od = out + dn * FF + lane;
#pragma unroll
  for (int f = 0; f < FF; f += 32) atomicAdd(od + f, coef * hs[f]);
}

__global__ __launch_bounds__(256) void aggregate_gen(
    const int* __restrict__ ei, const float* __restrict__ xb,
    const float* __restrict__ z, const float* __restrict__ h,
    float* __restrict__ out, int E, int Etot, int F) {
  const int wid = blockIdx.x * 8 + (threadIdx.x >> 5);
  if (wid >= Etot) return;
  const int lane = threadIdx.x & 31;
  int sn, dn;
  if (wid < E) { sn = ei[wid]; dn = ei[E + wid]; } else { sn = dn = wid - E; }
  const float coef = xb[wid] / z[dn];
  const float* __restrict__ hs = h + sn * F;
  float* __restrict__ od = out + dn * F;
  for (int f = lane; f < F; f += 32) atomicAdd(&od[f], coef * hs[f]);
}

// ---------------------------------------------------------------------------
// Orchestration
// ---------------------------------------------------------------------------
extern "C" void kernel_launch(void* const* d_in, const int* in_sizes, int n_in,
                              void* d_out, int out_size, void* d_ws,
                              size_t ws_size, hipStream_t stream) {
  const float* x   = (const float*)d_in[0];
  const int*   ei  = (const int*)d_in[1];
  const float* W1  = (const float*)d_in[2];
  const float* as1 = (const float*)d_in[3];
  const float* ad1 = (const float*)d_in[4];
  const float* b1  = (const float*)d_in[5];
  const float* W2  = (const float*)d_in[6];
  const float* as2 = (const float*)d_in[7];
  const float* ad2 = (const float*)d_in[8];
  const float* b2  = (const float*)d_in[9];
  float* out = (float*)d_out;

  const int H    = in_sizes[3];            // 256
  const int Fout = in_sizes[7];            // 128
  const int Fin  = in_sizes[2] / H;        // 128
  const int N    = in_sizes[0] / Fin;      // 50000
  const int E    = in_sizes[1] / 2;        // 800000
  const int Etot = E + N;                  // + self loops

  // Workspace carve-up (floats)
  float* ws = (float*)d_ws;
  float* h1 = ws;                     // N*H
  float* o1 = h1 + (size_t)N * H;     // N*H
  float* h2 = o1 + (size_t)N * H;     // N*Fout
  float* sb = h2 + (size_t)N * Fout;  // N
  float* db = sb + N;                 // N
  float* mx = db + N;                 // N
  float* zz = mx + N;                 // N
  float* eb = zz + N;                 // Etot
  float* xb = eb + Etot;              // Etot

  const int T = 256;
  const int nodeBlk  = (N + T - 1) / T;
  const int nodeWblk = (N + 7) / 8;       // wave-per-node
  const int edgeBlk  = (Etot + T - 1) / T;
  const int edgeWblk = (Etot + 7) / 8;    // wave-per-edge

  // ---------------- Layer 1 ----------------
  {
    const int waves = ((N + 15) / 16) * (H / 16);
    if (Fin == 128 && H == 256)
      gemm_wmma_fix<false, 256, 128><<<(waves + 7) / 8, T, 0, stream>>>(x, W1, h1, N);
    else
      gemm_wmma_gen<false><<<(waves + 7) / 8, T, 0, stream>>>(x, W1, h1, N, H, Fin);
    node_proj<<<nodeWblk, T, 0, stream>>>(h1, as1, ad1, sb, db, N, H);
    init_node<<<nodeBlk, T, 0, stream>>>(mx, zz, N);
    edge_max_k<<<edgeBlk, T, 0, stream>>>(ei, sb, db, eb, mx, E, Etot);
    edge_exp_k<<<edgeBlk, T, 0, stream>>>(ei, eb, mx, xb, zz, E, Etot);
    init_out_k<<<(N * H + T - 1) / T, T, 0, stream>>>(o1, b1, N * H, H);
    if (H == 256)
      aggregate_fix<256><<<edgeWblk, T, 0, stream>>>(ei, xb, zz, h1, o1, E, Etot);
    else
      aggregate_gen<<<edgeWblk, T, 0, stream>>>(ei, xb, zz, h1, o1, E, Etot, H);
  }

  // ---------------- Layer 2 (ReLU fused into GEMM A-load) ----------------
  {
    const int waves = ((N + 15) / 16) * (Fout / 16);
    if (H == 256 && Fout == 128)
      gemm_wmma_fix<true, 128, 256><<<(waves + 7) / 8, T, 0, stream>>>(o1, W2, h2, N);
    else
      gemm_wmma_gen<true><<<(waves + 7) / 8, T, 0, stream>>>(o1, W2, h2, N, Fout, H);
    node_proj<<<nodeWblk, T, 0, stream>>>(h2, as2, ad2, sb, db, N, Fout);
    init_node<<<nodeBlk, T, 0, stream>>>(mx, zz, N);
    edge_max_k<<<edgeBlk, T, 0, stream>>>(ei, sb, db, eb, mx, E, Etot);
    edge_exp_k<<<edgeBlk, T, 0, stream>>>(ei, eb, mx, xb, zz, E, Etot);
    init_out_k<<<(N * Fout + T - 1) / T, T, 0, stream>>>(out, b2, N * Fout, Fout);
    if (Fout == 128)
      aggregate_fix<128><<<edgeWblk, T, 0, stream>>>(ei, xb, zz, h2, out, E, Etot);
    else
      aggregate_gen<<<edgeWblk, T, 0, stream>>>(ei, xb, zz, h2, out, E, Etot, Fout);
  }
}